// Classifier_16183436772110
// MI455X (gfx1250) — compile-verified
//
#include <hip/hip_runtime.h>

// ---------------- sizes (fixed by the reference) ----------------
#define F_IN 128
#define H1   256
#define H2   128
#define HD   64
#define P    3
#define CLS  2

typedef __attribute__((ext_vector_type(2))) float v2f;
typedef __attribute__((ext_vector_type(8))) float v8f;

// Hardware fp32 atomic add (lowers to global_atomic_add_f32, no CAS loop)
__device__ __forceinline__ void atomic_add_f32(float* p, float v) {
    __hip_atomic_fetch_add(p, v, __ATOMIC_RELAXED, __HIP_MEMORY_SCOPE_AGENT);
}

// ---------------------------------------------------------------
// degree: deg[dst[e]] += 1
__global__ void degree_kernel(const int* __restrict__ dst, float* __restrict__ deg, int E) {
    int e = blockIdx.x * blockDim.x + threadIdx.x;
    if (e < E) atomic_add_f32(&deg[dst[e]], 1.0f);
}

// dinv = rsqrt(deg + 1)  (self loop adds 1; deg+1 > 0 always)
__global__ void dinv_kernel(float* __restrict__ deg, int n) {
    int i = blockIdx.x * blockDim.x + threadIdx.x;
    if (i < n) deg[i] = rsqrtf(deg[i] + 1.0f);
}

// ---------------------------------------------------------------
// WMMA fp32 GEMM: out[nrows,COUT] = A[nrows,KDIM] @ W[KDIM,COUT] (+bias)
// Block = 128 threads = 4 waves. Each block: 16 rows; wave w does a 16-col tile.
// A tile staged in LDS via b128; B streamed from L2-resident weights.
template<int KDIM, int COUT, bool ADD_BIAS>
__global__ void wmma_gemm_kernel(const float* __restrict__ A,
                                 const float* __restrict__ W,
                                 const float* __restrict__ bias,
                                 float* __restrict__ out, int nrows) {
    __shared__ __align__(16) float tileA[16 * KDIM];

    const int tid  = threadIdx.x;
    const int row0 = blockIdx.x * 16;
    const bool full = (row0 + 16 <= nrows);   // wave-uniform

    if (full) {
        // vectorized stage: 16*KDIM/4 float4's across 128 threads
        const float4* Ab = reinterpret_cast<const float4*>(A + (size_t)row0 * KDIM);
        float4* Tb = reinterpret_cast<float4*>(tileA);
        #pragma unroll
        for (int idx = tid; idx < 16 * KDIM / 4; idx += 128) Tb[idx] = Ab[idx];
    } else {
        for (int idx = tid; idx < 16 * KDIM; idx += 128) {
            int r = idx / KDIM, k = idx % KDIM;
            int gr = row0 + r;
            tileA[idx] = (gr < nrows) ? A[(size_t)gr * KDIM + k] : 0.0f;
        }
    }
    __syncthreads();

    const int wave = tid >> 5;
    const int lane = tid & 31;
    const int half = lane >> 4;    // 0: K pair (k,k+1)   1: K pair (k+2,k+3)
    const int l    = lane & 15;    // A: M index, B: N index
    const int n0   = blockIdx.y * (4 * 16) + wave * 16;

    v8f acc = {0.f, 0.f, 0.f, 0.f, 0.f, 0.f, 0.f, 0.f};

    #pragma unroll 4
    for (int k = 0; k < KDIM; k += 4) {
        const int ka = k + 2 * half;
        // A fragment (16x4 f32): lane l holds A[M=l][ka], A[M=l][ka+1]
        v2f a;
        a.x = tileA[l * KDIM + ka];
        a.y = tileA[l * KDIM + ka + 1];
        // B fragment (4x16 f32): lane l holds W[ka][n0+l], W[ka+1][n0+l]
        const float* wp = W + (size_t)ka * COUT + n0 + l;
        v2f b;
        b.x = wp[0];
        b.y = wp[COUT];
        acc = __builtin_amdgcn_wmma_f32_16x16x4_f32(
            /*neg_a=*/false, a, /*neg_b=*/false, b,
            /*c_mod=*/(short)0, acc, /*reuse_a=*/false, /*reuse_b=*/false);
    }

    // C/D layout: VGPR r -> (M=row0+r+8*half, N=n0+l)
    const int col = n0 + l;
    const float bv = ADD_BIAS ? bias[col] : 0.0f;
    float* op = out + (size_t)(row0 + 8 * half) * COUT + col;
    if (full) {
        #pragma unroll
        for (int r = 0; r < 8; ++r) op[(size_t)r * COUT] = acc[r] + bv;
    } else {
        #pragma unroll
        for (int r = 0; r < 8; ++r) {
            int row = row0 + r + 8 * half;
            if (row < nrows) out[(size_t)row * COUT + col] = acc[r] + bv;
        }
    }
}

// ---------------------------------------------------------------
// agg[dst] += h[src] * (dinv[src]*dinv[dst])
// H/4 threads per edge (float4 gather + 4 hw atomics); 256-thread blocks.
template<int H>
__global__ void aggregate_kernel(const float* __restrict__ h,
                                 const int* __restrict__ src,
                                 const int* __restrict__ dst,
                                 const float* __restrict__ dinv,
                                 float* __restrict__ agg, int E) {
    constexpr int TPE = H / 4;          // threads per edge
    constexpr int EPB = 256 / TPE;      // edges per block
    const int t  = threadIdx.x;
    const int c4 = t % TPE;
    const long e = (long)blockIdx.x * EPB + (t / TPE);
    if (e >= E) return;
    const int s = src[e];
    const int d = dst[e];
    const float norm = dinv[s] * dinv[d];
    const float4 hv = reinterpret_cast<const float4*>(h + (size_t)s * H)[c4];
    float* ap = agg + (size_t)d * H + (size_t)c4 * 4;
    atomic_add_f32(ap + 0, hv.x * norm);
    atomic_add_f32(ap + 1, hv.y * norm);
    atomic_add_f32(ap + 2, hv.z * norm);
    atomic_add_f32(ap + 3, hv.w * norm);
}

// agg = maybe_relu(agg + hpre*dinv^2 + bias)   (in place, float4)
template<int H, bool RELU>
__global__ void finalize_kernel(float* __restrict__ agg,
                                const float* __restrict__ hpre,
                                const float* __restrict__ dinv,
                                const float* __restrict__ bias,
                                int n) {
    constexpr int H4 = H / 4;
    const long idx4 = (long)blockIdx.x * blockDim.x + threadIdx.x;
    if (idx4 >= (long)n * H4) return;
    const int i  = (int)(idx4 / H4);
    const int c4 = (int)(idx4 % H4);
    const float di = dinv[i];
    const float sc = di * di;
    float4 a = reinterpret_cast<float4*>(agg)[idx4];
    const float4 p = reinterpret_cast<const float4*>(hpre)[idx4];
    const float4 b = reinterpret_cast<const float4*>(bias)[c4];
    a.x = a.x + p.x * sc + b.x;
    a.y = a.y + p.y * sc + b.y;
    a.z = a.z + p.z * sc + b.z;
    a.w = a.w + p.w * sc + b.w;
    if (RELU) {
        a.x = fmaxf(a.x, 0.f); a.y = fmaxf(a.y, 0.f);
        a.z = fmaxf(a.z, 0.f); a.w = fmaxf(a.w, 0.f);
    }
    reinterpret_cast<float4*>(agg)[idx4] = a;
}

// ---------------------------------------------------------------
// node_x[i,:] = hd[i,:] @ Wn[64,2] + bn    (float4 row walk)
__global__ void node_out_kernel(const float* __restrict__ hd,
                                const float* __restrict__ Wn,
                                const float* __restrict__ bn,
                                float* __restrict__ out, int n) {
    __shared__ float w[HD * CLS];
    for (int i = threadIdx.x; i < HD * CLS; i += blockDim.x) w[i] = Wn[i];
    __syncthreads();
    int i = blockIdx.x * blockDim.x + threadIdx.x;
    if (i >= n) return;
    float a0 = bn[0], a1 = bn[1];
    const float4* hp = reinterpret_cast<const float4*>(hd + (size_t)i * HD);
    #pragma unroll
    for (int k4 = 0; k4 < HD / 4; ++k4) {
        const float4 v = hp[k4];
        const float* wr = &w[k4 * 8];
        a0 += v.x * wr[0] + v.y * wr[2] + v.z * wr[4] + v.w * wr[6];
        a1 += v.x * wr[1] + v.y * wr[3] + v.z * wr[5] + v.w * wr[7];
    }
    out[(size_t)i * 2]     = a0;
    out[(size_t)i * 2 + 1] = a1;
}

// edge_x[e,:] = [hd[src], hd[dst], props] @ We[131,2] + be
__global__ void edge_out_kernel(const float* __restrict__ hd,
                                const int* __restrict__ src,
                                const int* __restrict__ dst,
                                const float* __restrict__ props,
                                const float* __restrict__ We,
                                const float* __restrict__ be,
                                float* __restrict__ out, int E) {
    __shared__ float w[(2 * HD + P) * CLS];
    for (int i = threadIdx.x; i < (2 * HD + P) * CLS; i += blockDim.x) w[i] = We[i];
    __syncthreads();
    int e = blockIdx.x * blockDim.x + threadIdx.x;
    if (e >= E) return;
    const int s = src[e];
    const int d = dst[e];
    const float4* hs = reinterpret_cast<const float4*>(hd + (size_t)s * HD);
    const float4* hv = reinterpret_cast<const float4*>(hd + (size_t)d * HD);
    float a0 = be[0], a1 = be[1];
    #pragma unroll
    for (int k4 = 0; k4 < HD / 4; ++k4) {
        const float4 v = hs[k4];
        const float* wr = &w[k4 * 8];
        a0 += v.x * wr[0] + v.y * wr[2] + v.z * wr[4] + v.w * wr[6];
        a1 += v.x * wr[1] + v.y * wr[3] + v.z * wr[5] + v.w * wr[7];
    }
    #pragma unroll
    for (int k4 = 0; k4 < HD / 4; ++k4) {
        const float4 v = hv[k4];
        const float* wr = &w[HD * 2 + k4 * 8];
        a0 += v.x * wr[0] + v.y * wr[2] + v.z * wr[4] + v.w * wr[6];
        a1 += v.x * wr[1] + v.y * wr[3] + v.z * wr[5] + v.w * wr[7];
    }
    #pragma unroll
    for (int p = 0; p < P; ++p) {
        const float v = props[(size_t)e * P + p];
        a0 += v * w[(2 * HD + p) * 2];
        a1 += v * w[(2 * HD + p) * 2 + 1];
    }
    out[(size_t)e * 2]     = a0;
    out[(size_t)e * 2 + 1] = a1;
}

// ---------------------------------------------------------------
extern "C" void kernel_launch(void* const* d_in, const int* in_sizes, int n_in,
                              void* d_out, int out_size, void* d_ws, size_t ws_size,
                              hipStream_t stream) {
    const float* x     = (const float*)d_in[0];
    const int*   eidx  = (const int*)  d_in[1];
    const float* props = (const float*)d_in[2];
    const float* W1 = (const float*)d_in[3];
    const float* b1 = (const float*)d_in[4];
    const float* W2 = (const float*)d_in[5];
    const float* b2 = (const float*)d_in[6];
    const float* Wd = (const float*)d_in[7];
    const float* bd = (const float*)d_in[8];
    const float* Wn = (const float*)d_in[9];
    const float* bn = (const float*)d_in[10];
    const float* We = (const float*)d_in[11];
    const float* be = (const float*)d_in[12];

    const int N = in_sizes[0] / F_IN;
    const int E = in_sizes[1] / 2;
    const int* src = eidx;
    const int* dst = eidx + E;

    // workspace layout (fp32): dinv[N] | bufB[N*256] | bufC[N*256]
    float* ws   = (float*)d_ws;
    float* dinv = ws;
    float* bufB = ws + (((size_t)N + 255) & ~(size_t)255);
    float* bufC = bufB + (size_t)N * H1;

    float* node_out = (float*)d_out;
    float* edge_out = (float*)d_out + (size_t)N * CLS;

    // ---- degree / normalization ----
    hipMemsetAsync(dinv, 0, (size_t)N * sizeof(float), stream);
    degree_kernel<<<(E + 255) / 256, 256, 0, stream>>>(dst, dinv, E);
    dinv_kernel<<<(N + 255) / 256, 256, 0, stream>>>(dinv, N);

    // ---- layer 1: h1_pre = x @ W1  -> bufB ----
    {
        dim3 grid((N + 15) / 16, H1 / 64);
        wmma_gemm_kernel<F_IN, H1, false><<<grid, 128, 0, stream>>>(x, W1, nullptr, bufB, N);
    }
    hipMemsetAsync(bufC, 0, (size_t)N * H1 * sizeof(float), stream);
    {
        constexpr int EPB = 256 / (H1 / 4);
        aggregate_kernel<H1><<<(E + EPB - 1) / EPB, 256, 0, stream>>>(
            bufB, src, dst, dinv, bufC, E);
    }
    {
        long total4 = (long)N * (H1 / 4);
        finalize_kernel<H1, true><<<(unsigned)((total4 + 255) / 256), 256, 0, stream>>>(
            bufC, bufB, dinv, b1, N);
    }
    // bufC = h1

    // ---- layer 2: h2_pre = h1 @ W2 -> bufB ----
    {
        dim3 grid((N + 15) / 16, H2 / 64);
        wmma_gemm_kernel<H1, H2, false><<<grid, 128, 0, stream>>>(bufC, W2, nullptr, bufB, N);
    }
    hipMemsetAsync(bufC, 0, (size_t)N * H2 * sizeof(float), stream);
    {
        constexpr int EPB = 256 / (H2 / 4);
        aggregate_kernel<H2><<<(E + EPB - 1) / EPB, 256, 0, stream>>>(
            bufB, src, dst, dinv, bufC, E);
    }
    {
        long total4 = (long)N * (H2 / 4);
        finalize_kernel<H2, false><<<(unsigned)((total4 + 255) / 256), 256, 0, stream>>>(
            bufC, bufB, dinv, b2, N);
    }
    // bufC = h2

    // ---- dense: hd = h2 @ Wd + bd -> bufB ----
    {
        dim3 grid((N + 15) / 16, HD / 64);
        wmma_gemm_kernel<H2, HD, true><<<grid, 128, 0, stream>>>(bufC, Wd, bd, bufB, N);
    }

    // ---- heads ----
    node_out_kernel<<<(N + 255) / 256, 256, 0, stream>>>(bufB, Wn, bn, node_out, N);
    edge_out_kernel<<<(E + 255) / 256, 256, 0, stream>>>(bufB, src, dst, props, We, be,
                                                         edge_out, E);
}